// RBDispatcher_33535104647679
// MI455X (gfx1250) — compile-verified
//
#include <hip/hip_runtime.h>
#include <cstdint>

// Problem constants from the reference.
#define D_MODEL        2048
#define TOP_K          2
#define THREADS        256            // 8 waves (wave32) per block
#define ROWS_PER_BLOCK 4              // double-buffered 2-stage pipeline
// Each thread moves 2 x 16B chunks per row => 256*2*16B = 8192B = full row.

#define AS1 __attribute__((address_space(1)))   // global
#define AS3 __attribute__((address_space(3)))   // LDS

typedef int   v4i __attribute__((ext_vector_type(4)));
typedef float v4f __attribute__((ext_vector_type(4)));
typedef AS1 v4i* gvec_t;   // global 16B chunk pointer
typedef AS3 v4i* lvec_t;   // LDS    16B chunk pointer

#if __has_builtin(__builtin_amdgcn_global_load_async_to_lds_b128)
#define USE_ASYNC_LDS 1
#else
#define USE_ASYNC_LDS 0
#endif

// Fused routing chain (uniform per block -> scalar loads):
//   r <  n_s2 : src = idx_s1[s1_to_s2[r]] / top_k
//   r >= n_s2 : src = idx_s1[r - n_s2]    / top_k
__device__ __forceinline__ long src_row(const int* __restrict__ idx_s1,
                                        const int* __restrict__ s1_to_s2,
                                        int row, int n_s2)
{
    const int expanded = (row < n_s2) ? idx_s1[s1_to_s2[row]]
                                      : idx_s1[row - n_s2];
    return (long)((unsigned)expanded / TOP_K);   // shift, not idiv
}

#if USE_ASYNC_LDS

// Partial wait on ASYNCcnt. Only async *loads* live on this counter in this
// kernel, and loads complete in order, so waiting cnt<=N with N ops issued
// after a given load guarantees that load's LDS data is resident.
template <int N>
__device__ __forceinline__ void wait_async()
{
#if __has_builtin(__builtin_amdgcn_s_wait_asynccnt)
    __builtin_amdgcn_s_wait_asynccnt(N);
#else
    asm volatile("s_wait_asynccnt %0" :: "i"(N) : "memory");
#endif
    asm volatile("" ::: "memory");   // keep DS reads below the wait
}

// Issue one row's gather: global -> LDS via the CDNA5 async data mover.
// Each thread owns two disjoint 16B chunks (float4 #t and #(t+256)).
__device__ __forceinline__ void async_load_row(const float* __restrict__ x,
                                               long src, float* lbuf, int t)
{
    const float* srow = x + src * (long)D_MODEL;
    __builtin_amdgcn_global_load_async_to_lds_b128(
        (gvec_t)(srow + 4 * t),               (lvec_t)(lbuf + 4 * t),               0, 0);
    __builtin_amdgcn_global_load_async_to_lds_b128(
        (gvec_t)(srow + 4 * t + 4 * THREADS), (lvec_t)(lbuf + 4 * t + 4 * THREADS), 0, 0);
}

// Drain one row: LDS -> VGPR (ds_load_b128) -> nontemporal global store.
// NT stores keep the 192MB write-once output from evicting the L2-resident
// x table; STOREcnt is never waited on inside the loop (S_ENDPGM drains it).
__device__ __forceinline__ void store_row(const float* lbuf,
                                          float* __restrict__ out,
                                          int row, int t)
{
    const v4f* l4 = (const v4f*)lbuf;
    v4f a = l4[t];
    v4f b = l4[t + THREADS];
    v4f* d4 = (v4f*)(out + (long)row * (long)D_MODEL);
    __builtin_nontemporal_store(a, d4 + t);
    __builtin_nontemporal_store(b, d4 + t + THREADS);
}

#endif // USE_ASYNC_LDS

__global__ __launch_bounds__(THREADS)
void moe_dispatch_gather(const float* __restrict__ x,
                         const int*   __restrict__ idx_s1,
                         const int*   __restrict__ s1_to_s2,
                         float*       __restrict__ out,
                         int n_s2)
{
    const int t    = threadIdx.x;
    const int base = blockIdx.x * ROWS_PER_BLOCK;

#if USE_ASYNC_LDS
    __shared__ float lbuf[2][D_MODEL];     // 2 x 8KB double buffer

    // Prologue: fill both buffers (rows base+0, base+1 in flight).
    async_load_row(x, src_row(idx_s1, s1_to_s2, base + 0, n_s2), lbuf[0], t);
    async_load_row(x, src_row(idx_s1, s1_to_s2, base + 1, n_s2), lbuf[1], t);

    // Steady state: wait for the oldest row (<=2 leaves the prefetch in
    // flight), drain it, then immediately reuse its buffer for row k+2.
    wait_async<2>();                                   // row base+0 resident
    store_row(lbuf[0], out, base + 0, t);              // buf0 consumed to VGPRs
    async_load_row(x, src_row(idx_s1, s1_to_s2, base + 2, n_s2), lbuf[0], t);

    wait_async<2>();                                   // row base+1 resident
    store_row(lbuf[1], out, base + 1, t);
    async_load_row(x, src_row(idx_s1, s1_to_s2, base + 3, n_s2), lbuf[1], t);

    wait_async<2>();                                   // row base+2 resident
    store_row(lbuf[0], out, base + 2, t);

    wait_async<0>();                                   // row base+3 resident
    store_row(lbuf[1], out, base + 3, t);
    // S_ENDPGM's implicit wait-idle drains the outstanding NT stores.
#else
    // Portable fallback (also what the host pass parses): direct B128 copy.
    for (int k = 0; k < ROWS_PER_BLOCK; ++k) {
        const int  row = base + k;
        const long src = src_row(idx_s1, s1_to_s2, row, n_s2);
        const v4f* s4  = (const v4f*)(x   + src       * (long)D_MODEL);
        v4f*       d4  = (v4f*)      (out + (long)row * (long)D_MODEL);
        v4f a = s4[t];
        v4f b = s4[t + THREADS];
        __builtin_nontemporal_store(a, d4 + t);
        __builtin_nontemporal_store(b, d4 + t + THREADS);
    }
#endif
}

extern "C" void kernel_launch(void* const* d_in, const int* in_sizes, int n_in,
                              void* d_out, int out_size, void* d_ws, size_t ws_size,
                              hipStream_t stream)
{
    // setup_inputs() order: x, indices_s1, s1exp_to_s2_indices, n_tokens, top_k
    const float* x        = (const float*)d_in[0];
    const int*   idx_s1   = (const int*)  d_in[1];   // integer inputs -> int32 per harness
    const int*   s1_to_s2 = (const int*)  d_in[2];
    float*       out      = (float*)d_out;

    const int n_s1   = in_sizes[1];          // 16384 expanded rows
    const int n_s2   = in_sizes[2];          // 8192 re-gathered rows
    const int n_rows = n_s1 + n_s2;          // 24576 output rows (mult. of 4)

    moe_dispatch_gather<<<n_rows / ROWS_PER_BLOCK, THREADS, 0, stream>>>(
        x, idx_s1, s1_to_s2, out, n_s2);
}